// LightWrapper_28449863369107
// MI455X (gfx1250) — compile-verified
//
#include <hip/hip_runtime.h>
#include <math.h>

// ---------------------------------------------------------------------------
// IBL Monte-Carlo shading: P=131072 points, S=64 GGX-VNDF spec + 64 cosine
// diffuse samples, bilinear lat-long env lookups (512x1024x3 fp32).
//  - one wave32 per point; lane l handles samples l and l+32
//  - random arrays staged global->LDS via gfx1250 async copies (coalesced)
//  - env map pre-transposed planar->interleaved in d_ws: 4x b96 gathers per
//    bilinear sample instead of 12x b32 (3x fewer L2 requests)
//  - per-point attributes via readfirstlane -> scalar SMEM loads
//  - branchless fast atan2/acos (kernel is VALU-bound)
// ---------------------------------------------------------------------------

#define NSAMP 64
#define WPB   8              // waves (= points) per block
#define BLOCK (WPB * 32)
#define EM_H  512
#define EM_W  1024

#if defined(__gfx1250__) && \
    __has_builtin(__builtin_amdgcn_global_load_async_to_lds_b128) && \
    __has_builtin(__builtin_amdgcn_s_wait_asynccnt)
#define CDNA5_ASYNC 1
#else
#define CDNA5_ASYNC 0
#endif

#if CDNA5_ASYNC
typedef int v4i_t __attribute__((ext_vector_type(4)));
typedef __attribute__((address_space(1))) v4i_t* g_v4i_ptr;   // global v4i*
typedef __attribute__((address_space(3))) v4i_t* l_v4i_ptr;   // LDS v4i*
#endif

struct alignas(4) F3 { float x, y, z; };   // 12B interleaved texel (b96 load)

__device__ __forceinline__ float smith_g(float a2, float c) {
  return 2.0f * c / (c + sqrtf(a2 + (1.0f - a2) * c * c));
}

// Branchless atan2, minimax poly on [0,1]; max err ~1e-5 rad
// (~0.003 px on the 1024-wide env map -> negligible for bilinear MC sampling).
__device__ __forceinline__ float fast_atan2f(float y, float x) {
  float ax = fabsf(x), ay = fabsf(y);
  float mx = fmaxf(ax, ay), mn = fminf(ax, ay);
  float a = mn * __builtin_amdgcn_rcpf(fmaxf(mx, 1e-30f));
  float s = a * a;
  float r = fmaf(s, fmaf(s, fmaf(s, fmaf(s, 0.0208351f, -0.0851330f),
                                 0.1801410f), -0.3302995f), 0.9998660f) * a;
  r = (ay > ax) ? (1.5707963267948966f - r) : r;
  r = (x < 0.0f) ? (3.1415926535897932f - r) : r;
  return copysignf(r, y);
}

// Branchless acos: sqrt(1-|x|)*poly7(|x|) (A&S 4.4.46-style, err ~2e-8),
// reflected for negative inputs. Caller guarantees x in [-1,1].
__device__ __forceinline__ float fast_acosf(float x) {
  float xa = fabsf(x);
  float t = sqrtf(fmaxf(1.0f - xa, 0.0f));
  float p = fmaf(xa, fmaf(xa, fmaf(xa, fmaf(xa, fmaf(xa, fmaf(xa,
            fmaf(xa, -0.0012624911f, 0.0066700901f), -0.0170881256f),
            0.0308918810f), -0.0501743046f), 0.0889789874f),
            -0.2145988016f), 1.5707963050f);
  float r = t * p;
  return (x < 0.0f) ? (3.1415926535897932f - r) : r;
}

// Direction -> bilinear footprint (indices + weights), border-clamped.
__device__ __forceinline__ void env_coords(float dx, float dy, float dz,
                                           int& i00, int& i01, int& i10, int& i11,
                                           float& w00, float& w01, float& w10, float& w11) {
  const float INV_PI = 0.3183098861837907f;
  float u = 0.5f * (1.0f + fast_atan2f(dx, -dz) * INV_PI);
  float v = fast_acosf(fminf(fmaxf(dy, -1.0f), 1.0f)) * INV_PI;
  float xf = u * (float)EM_W - 0.5f;    // ((2u-1)+1)*W-1)*0.5
  float yf = v * (float)EM_H - 0.5f;
  float x0f = floorf(xf), y0f = floorf(yf);
  float wx = xf - x0f, wy = yf - y0f;
  int x0 = (int)x0f, y0 = (int)y0f;
  int x1 = x0 + 1, y1 = y0 + 1;
  x0 = max(0, min(x0, EM_W - 1)); x1 = max(0, min(x1, EM_W - 1));
  y0 = max(0, min(y0, EM_H - 1)); y1 = max(0, min(y1, EM_H - 1));
  i00 = y0 * EM_W + x0; i01 = y0 * EM_W + x1;
  i10 = y1 * EM_W + x0; i11 = y1 * EM_W + x1;
  w00 = (1.0f - wx) * (1.0f - wy);
  w01 = wx * (1.0f - wy);
  w10 = (1.0f - wx) * wy;
  w11 = wx * wy;
}

template <bool INTERLEAVED>
__device__ __forceinline__ void env_bilinear(const float* __restrict__ tex,
                                             float dx, float dy, float dz,
                                             float& r0, float& r1, float& r2) {
  int i00, i01, i10, i11;
  float w00, w01, w10, w11;
  env_coords(dx, dy, dz, i00, i01, i10, i11, w00, w01, w10, w11);
  if (INTERLEAVED) {
    const F3* t = (const F3*)tex;
    F3 c00 = t[i00], c01 = t[i01], c10 = t[i10], c11 = t[i11];
    r0 = w00 * c00.x + w01 * c01.x + w10 * c10.x + w11 * c11.x;
    r1 = w00 * c00.y + w01 * c01.y + w10 * c10.y + w11 * c11.y;
    r2 = w00 * c00.z + w01 * c01.z + w10 * c10.z + w11 * c11.z;
  } else {
    const float* c0 = tex;
    const float* c1 = tex + EM_H * EM_W;
    const float* c2 = tex + 2 * EM_H * EM_W;
    r0 = w00 * c0[i00] + w01 * c0[i01] + w10 * c0[i10] + w11 * c0[i11];
    r1 = w00 * c1[i00] + w01 * c1[i01] + w10 * c1[i10] + w11 * c1[i11];
    r2 = w00 * c2[i00] + w01 * c2[i01] + w10 * c2[i10] + w11 * c2[i11];
  }
}

// planar [3][H][W] -> interleaved [H][W][3]
__global__ __launch_bounds__(256) void transpose_emap(const float* __restrict__ src,
                                                      float* __restrict__ dst, int n) {
  int i = blockIdx.x * 256 + threadIdx.x;
  if (i < n) {
    float r = src[i], g = src[n + i], b = src[2 * n + i];
    dst[3 * i + 0] = r;
    dst[3 * i + 1] = g;
    dst[3 * i + 2] = b;
  }
}

template <bool INTERLEAVED>
__global__ __launch_bounds__(BLOCK) void ibl_kernel(
    const float* __restrict__ tex, const float* __restrict__ view_dir,
    const float* __restrict__ normal, const float* __restrict__ albedo,
    const float* __restrict__ rough, const float* __restrict__ metal,
    const float* __restrict__ u1, const float* __restrict__ u2,
    const float* __restrict__ uc, const float* __restrict__ vc,
    float* __restrict__ out, int P) {
  // rnd[array 0..3][wave 0..7][sample 0..63]  = 8 KB
  __shared__ float rnd[4 * WPB * NSAMP];

  const int t    = threadIdx.x;
  const int wave = t >> 5;
  const int lane = t & 31;
  const int p0   = blockIdx.x * WPB;
  // wave-uniform point index, made explicit for the compiler -> SMEM loads
  const int p    = __builtin_amdgcn_readfirstlane(p0 + wave);

  // ---- stage random numbers into LDS; wave pair owns one array; selector is
  // ---- readfirstlane'd so the pointer select is a scalar s_cselect.
  if (p0 + WPB <= P) {
    const int a = __builtin_amdgcn_readfirstlane(wave >> 1);   // 0..3
    const float* src = (a == 0) ? u1 : (a == 1) ? u2 : (a == 2) ? uc : vc;
#pragma unroll
    for (int k = 0; k < 2; ++k) {
      int f4 = (wave & 1) * 64 + k * 32 + lane;                // float4 idx [0,128)
      const float* g = src + (size_t)p0 * NSAMP + (size_t)f4 * 4;
      int q = a * 128 + f4;                                    // LDS float4 slot
#if CDNA5_ASYNC
      __builtin_amdgcn_global_load_async_to_lds_b128(
          (g_v4i_ptr)(uintptr_t)g, (l_v4i_ptr)&rnd[q * 4], 0, 0);
#else
      *(float4*)(&rnd[q * 4]) = *(const float4*)g;
#endif
    }
#if CDNA5_ASYNC
    __builtin_amdgcn_s_wait_asynccnt(0);
#endif
  } else {
    // partial tail block (not hit for P=131072): guarded scalar staging
    for (int q = t; q < 4 * WPB * NSAMP; q += BLOCK) {
      int a = q / (WPB * NSAMP);
      int off = q - a * (WPB * NSAMP);
      const float* src = (a == 0) ? u1 : (a == 1) ? u2 : (a == 2) ? uc : vc;
      size_t fidx = (size_t)p0 * NSAMP + off;
      rnd[a * (WPB * NSAMP) + off] = (fidx < (size_t)P * NSAMP) ? src[fidx] : 0.0f;
    }
  }
  __syncthreads();
  if (p >= P) return;

  // ----------- per-point setup (uniform p -> scalar SMEM loads) -----------
  float vx = view_dir[3 * p + 0], vy = view_dir[3 * p + 1], vz = view_dir[3 * p + 2];
  float nx = normal[3 * p + 0],  ny = normal[3 * p + 1],  nz = normal[3 * p + 2];
  float ar = albedo[3 * p + 0],  ag = albedo[3 * p + 1],  ab = albedo[3 * p + 2];
  float rg = rough[p], mt = metal[p];
  float alpha = rg * rg;
  float al2   = alpha * alpha;

  // tangent = normalize(cross((0,1,0), n)) = normalize((nz,0,-nx)); degen -> (1,0,0)
  float d2 = nx * nx + nz * nz;
  float tx, ty, tz;
  if (d2 > 0.0f) {
    float inv = rsqrtf(d2);
    tx = nz * inv; ty = 0.0f; tz = -nx * inv;
  } else {
    tx = 1.0f; ty = 0.0f; tz = 0.0f;
  }
  // bitangent = normalize(cross(n, t))
  float bx = ny * tz - nz * ty, by = nz * tx - nx * tz, bz = nx * ty - ny * tx;
  {
    float inv = rsqrtf(fmaxf(bx * bx + by * by + bz * bz, 1e-24f));
    bx *= inv; by *= inv; bz *= inv;
  }
  // v_local = TBN * v
  float vlx = tx * vx + ty * vy + tz * vz;
  float vly = bx * vx + by * vy + bz * vz;
  float vlz = nx * vx + ny * vy + nz * vz;

  float ndv   = fminf(fmaxf(nx * vx + ny * vy + nz * vz, 0.0f), 1.0f);
  float gview = smith_g(al2, ndv);
  float inv_sview = 1.0f / fminf(fmaxf(gview, 1e-10f), 1.0f);

  float f0r = mt * ar + (1.0f - mt) * 0.04f;
  float f0g = mt * ag + (1.0f - mt) * 0.04f;
  float f0b = mt * ab + (1.0f - mt) * 0.04f;
  float kdr = ar * (1.0f - mt), kdg = ag * (1.0f - mt), kdb = ab * (1.0f - mt);

  // VNDF frame is sample-invariant: hoist out of the sample loop.
  float vhx = vlx * alpha, vhy = vly * alpha, vhz = vlz;
  {
    float inv = rsqrtf(fmaxf(vhx * vhx + vhy * vhy + vhz * vhz, 1e-24f));
    vhx *= inv; vhy *= inv; vhz *= inv;
  }
  float l2 = vhx * vhx + vhy * vhy;
  float b1x, b1y;                                   // b1z == 0
  if (l2 > 0.0f) {
    float inv = rsqrtf(l2);
    b1x = -vhy * inv; b1y = vhx * inv;
  } else {
    b1x = 1.0f; b1y = 0.0f;
  }
  float b2x = -vhz * b1y, b2y = vhz * b1x, b2z = vhx * b1y - vhy * b1x;
  {
    float inv = rsqrtf(fmaxf(b2x * b2x + b2y * b2y + b2z * b2z, 1e-24f));
    b2x *= inv; b2y *= inv; b2z *= inv;
  }
  float sfac = 0.5f * (1.0f + vhz);

  const float TWO_PI = 6.283185307179586f;
  float accr = 0.0f, accg = 0.0f, accb = 0.0f;

  const float* rnd_u1 = &rnd[(0 * WPB + wave) * NSAMP];
  const float* rnd_u2 = &rnd[(1 * WPB + wave) * NSAMP];
  const float* rnd_uc = &rnd[(2 * WPB + wave) * NSAMP];
  const float* rnd_vc = &rnd[(3 * WPB + wave) * NSAMP];

#pragma unroll
  for (int k = 0; k < 2; ++k) {
    const int s = lane + k * 32;

    // ---------------- specular (GGX VNDF) ----------------
    {
      float U1 = rnd_u1[s], U2 = rnd_u2[s];
      float r = sqrtf(U1);
      float sth, cth;
      __sincosf(TWO_PI * U2, &sth, &cth);
      float t1 = r * cth;
      float t2 = r * sth;
      t2 = (1.0f - sfac) * sqrtf(fmaxf(1.0f - t1 * t1, 0.0f)) + sfac * t2;
      float t3 = sqrtf(fminf(fmaxf(1.0f - t1 * t1 - t2 * t2, 0.0f), 1.0f));
      float mhx = t1 * b1x + t2 * b2x + t3 * vhx;
      float mhy = t1 * b1y + t2 * b2y + t3 * vhy;
      float mhz =             t2 * b2z + t3 * vhz;   // b1z == 0
      mhx *= alpha; mhy *= alpha;                    // m_local = normalize(mh*a3)
      {
        float inv = rsqrtf(fmaxf(mhx * mhx + mhy * mhy + mhz * mhz, 1e-24f));
        mhx *= inv; mhy *= inv; mhz *= inv;
      }
      // local -> world (TBN^T)
      float mwx = mhx * tx + mhy * bx + mhz * nx;
      float mwy = mhx * ty + mhy * by + mhz * ny;
      float mwz = mhx * tz + mhy * bz + mhz * nz;
      // light = normalize(2*(v.m)*m - v)
      float idn = vx * mwx + vy * mwy + vz * mwz;
      float lx = 2.0f * idn * mwx - vx;
      float ly = 2.0f * idn * mwy - vy;
      float lz = 2.0f * idn * mwz - vz;
      {
        float inv = rsqrtf(fmaxf(lx * lx + ly * ly + lz * lz, 1e-24f));
        lx *= inv; ly *= inv; lz *= inv;
      }
      float er, eg, eb;
      env_bilinear<INTERLEAVED>(tex, lx, ly, lz, er, eg, eb);
      // half vector terms
      float hx = vx + lx, hy = vy + ly, hz = vz + lz;
      {
        float inv = rsqrtf(fmaxf(hx * hx + hy * hy + hz * hz, 1e-24f));
        hx *= inv; hy *= inv; hz *= inv;
      }
      float hdv = fminf(fmaxf(hx * vx + hy * vy + hz * vz, 0.0f), 1.0f);
      float ndl = fminf(fmaxf(nx * lx + ny * ly + nz * lz, 0.0f), 1.0f);
      float geo = gview * smith_g(al2, ndl);
      float omh = 1.0f - hdv;
      float omh2 = omh * omh;
      float f5 = omh2 * omh2 * omh;
      float kfac = geo * inv_sview;
      accr += (f0r + (1.0f - f0r) * f5) * er * kfac;
      accg += (f0g + (1.0f - f0g) * f5) * eg * kfac;
      accb += (f0b + (1.0f - f0b) * f5) * eb * kfac;
    }

    // ---------------- diffuse (cosine hemisphere) ----------------
    {
      float UC = rnd_uc[s], VC = rnd_vc[s];
      float sph, cph;
      __sincosf(TWO_PI * UC, &sph, &cph);
      float ct = sqrtf(VC);
      float st = sqrtf(fmaxf(1.0f - VC, 0.0f));
      float dxl = cph * st, dyl = sph * st, dzl = ct;
      float wxd = dxl * tx + dyl * bx + dzl * nx;
      float wyd = dxl * ty + dyl * by + dzl * ny;
      float wzd = dxl * tz + dyl * bz + dzl * nz;
      float er, eg, eb;
      env_bilinear<INTERLEAVED>(tex, wxd, wyd, wzd, er, eg, eb);
      accr += er * kdr;
      accg += eg * kdg;
      accb += eb * kdb;
    }
  }

  // ---- wave32 butterfly reduction over the 64 samples ----
#pragma unroll
  for (int m = 16; m >= 1; m >>= 1) {
    accr += __shfl_xor(accr, m, 32);
    accg += __shfl_xor(accg, m, 32);
    accb += __shfl_xor(accb, m, 32);
  }
  if (lane == 0) {
    const float scale = 1.0f / (float)NSAMP;
    out[3 * p + 0] = accr * scale;
    out[3 * p + 1] = accg * scale;
    out[3 * p + 2] = accb * scale;
  }
}

extern "C" void kernel_launch(void* const* d_in, const int* in_sizes, int n_in,
                              void* d_out, int out_size, void* d_ws, size_t ws_size,
                              hipStream_t stream) {
  const float* emap      = (const float*)d_in[0];
  const float* view_dir  = (const float*)d_in[1];
  const float* normal    = (const float*)d_in[2];
  const float* albedo    = (const float*)d_in[3];
  const float* roughness = (const float*)d_in[4];
  const float* metalness = (const float*)d_in[5];
  const float* u1        = (const float*)d_in[6];
  const float* u2        = (const float*)d_in[7];
  const float* uc        = (const float*)d_in[8];
  const float* vc        = (const float*)d_in[9];
  float* out = (float*)d_out;

  const int P = in_sizes[1] / 3;          // view_dir is P x 3
  const int blocks = (P + WPB - 1) / WPB; // one wave32 per point, 8 per block
  const int npix = EM_H * EM_W;
  const bool interleave = ws_size >= (size_t)3 * npix * sizeof(float);

  if (interleave) {
    float* tex = (float*)d_ws;
    hipLaunchKernelGGL(transpose_emap, dim3((npix + 255) / 256), dim3(256), 0,
                       stream, emap, tex, npix);
    hipLaunchKernelGGL((ibl_kernel<true>), dim3(blocks), dim3(BLOCK), 0, stream,
                       tex, view_dir, normal, albedo, roughness, metalness,
                       u1, u2, uc, vc, out, P);
  } else {
    hipLaunchKernelGGL((ibl_kernel<false>), dim3(blocks), dim3(BLOCK), 0, stream,
                       emap, view_dir, normal, albedo, roughness, metalness,
                       u1, u2, uc, vc, out, P);
  }
}